// CAM_Module_6365141532876
// MI455X (gfx1250) — compile-verified
//
#include <hip/hip_runtime.h>
#include <hip/hip_bf16.h>

// ---------------------------------------------------------------------------
// CAM module fused kernel for MI455X (gfx1250).
//
// x: (B=8, C=256, P=3, H=64, W=64) f32.  For each (b,c): V = 3x4096 tile.
//   energy = V V^T (3x3)  ->  attn = softmax(rowmax - energy)
//   out    = gamma * (attn V) + x
//
// Bandwidth-bound (AI ~= 3 FLOP/B, HBM floor ~8.6us): single-pass design.
//  * TDM tensor_load_to_lds stages the contiguous 48KB tile once (TENSORcnt).
//  * v_wmma_f32_16x16x4_f32 computes the Gram matrix via block-diagonal
//    packing (5 chunk-pairs per WMMA, symmetric => A and B reg images equal).
//  * Non-temporal b128 stores for the streamed output.
// ---------------------------------------------------------------------------

typedef __attribute__((ext_vector_type(2))) float v2f;
typedef __attribute__((ext_vector_type(4))) float v4f;
typedef __attribute__((ext_vector_type(8))) float v8f;
typedef __attribute__((ext_vector_type(4))) unsigned int v4u;
typedef __attribute__((ext_vector_type(4))) int v4i;
typedef __attribute__((ext_vector_type(8))) int v8i;

#define HW   4096
#define PN   3
#define TILE (PN * HW)           // 12288 floats = 48 KB

__global__ __launch_bounds__(256)
void cam_fused_kernel(const float* __restrict__ x,
                      const float* __restrict__ gamma,
                      float* __restrict__ out)
{
    __shared__ float ldsV[TILE];                       // 48 KB staged tile
    __shared__ float ldsD[8 * 256];                    // per-wave 16x16 D dumps
    __shared__ float ldsGram[6];
    __shared__ float ldsAtt[9];
    __shared__ __attribute__((aligned(8))) float ldsZ[2]; // zero slot for padding lanes

    const int tid  = threadIdx.x;
    const int wave = tid >> 5;
    const int lane = tid & 31;
    const unsigned long long base = (unsigned long long)blockIdx.x * TILE;

    // ---- Phase 0: TDM bulk DMA  global -> LDS  (48 KB, 1D tile) ----------
    if (tid == 0) {
        ldsZ[0] = 0.0f; ldsZ[1] = 0.0f;

        // Low 32 bits of a generic LDS pointer == LDS byte offset (flat
        // aperture mapping).
        unsigned ldsOff = (unsigned)(unsigned long long)(&ldsV[0]);
        unsigned long long ga = (unsigned long long)(x + base);

        // D# group 0: count=1 | lds_addr | global_addr | type=2 ("image")
        v4u g0;
        g0[0] = 1u;
        g0[1] = ldsOff;
        g0[2] = (unsigned)ga;
        g0[3] = ((unsigned)(ga >> 32) & 0x01FFFFFFu) | (2u << 30);

        // D# group 1: data_size=4B; tensor_dim0 = tile_dim0 = 12288;
        //             tensor_dim1 = 1; tensor_dim0_stride = 12288.
        v8i g1;
        g1[0] = (int)(2u << 16);            // data_size = 2 (4 bytes)
        g1[1] = (int)(0x3000u << 16);       // tensor_dim0[15:0] = 12288
        g1[2] = (int)(1u << 16);            // tensor_dim0 hi=0 | tensor_dim1 lo=1
        g1[3] = (int)(0x3000u << 16);       // tensor_dim1 hi=0 | tile_dim0 = 12288
        g1[4] = 1;                          // tile_dim1 = 1
        g1[5] = 12288;                      // tensor_dim0_stride lo32
        g1[6] = 0;
        g1[7] = 0;

        v4i g2 = {0, 0, 0, 0};
        v4i g3 = {0, 0, 0, 0};
        v8i g4 = {0, 0, 0, 0, 0, 0, 0, 0};  // 6-arg (clang-23) form: extra group

        __builtin_amdgcn_tensor_load_to_lds(g0, g1, g2, g3, g4, 0);
        __builtin_amdgcn_s_wait_tensorcnt(0);   // issuing wave drains TENSORcnt
    }
    __syncthreads();                             // LDS tile visible to all waves

    // ---- Phase 1: Gram matrix via v_wmma_f32_16x16x4_f32 -----------------
    // A rows m=0..14: v[m%3][kbase + (m/3)*4 .. +3]; row 15 = 0.
    // Symmetric product => B register image identical to A's; one ds_load_b64
    // per lane fills both operand VGPRs. Each WMMA covers 20 K-elements;
    // wave w owns K in [w*512, w*512+512) -> 26 self-accumulating WMMAs.
    const int m      = lane & 15;
    const int khalf  = lane >> 4;                // 0: K0/K1, 1: K2/K3
    const int row    = m % 3;
    const int chunk  = m / 3;                    // 0..4 valid, m==15 padded
    const bool vlane = (m != 15);
    const int kstart = wave * 512;
    const int kend   = kstart + 512;
    const int e      = chunk * 4 + khalf * 2;

    v8f acc = {};
    for (int it = 0; it < 26; ++it) {
        const int kbase  = kstart + it * 20;
        const int cstart = kbase + chunk * 4;
        const bool in    = vlane && (cstart < kend);
        const float* src = in ? &ldsV[row * HW + kbase + e] : &ldsZ[0];
        v2f a = *(const v2f*)src;
        acc = __builtin_amdgcn_wmma_f32_16x16x4_f32(
            false, a, false, a, (short)0, acc, false, false);
    }

    // Dump D (16x16) to LDS: VGPR p, lane L -> D[p + khalf*8][L&15].
    #pragma unroll
    for (int p = 0; p < 8; ++p) {
        ldsD[wave * 256 + (p + khalf * 8) * 16 + m] = acc[p];
    }
    __syncthreads();

    // Reduce the 5 diagonal 3x3 blocks across 8 waves -> 6 unique Gram vals.
    if (tid < 6) {
        const int i = (tid < 3) ? 0 : ((tid < 5) ? 1 : 2);
        const int j = (tid < 3) ? tid : ((tid < 5) ? (tid - 2) : 2);
        float s = 0.0f;
        for (int w = 0; w < 8; ++w)
            #pragma unroll
            for (int c = 0; c < 5; ++c)
                s += ldsD[w * 256 + (3 * c + i) * 16 + (3 * c + j)];
        ldsGram[tid] = s;
    }
    __syncthreads();

    // ---- Phase 2: 3x3 softmax(rowmax - energy) (scalar noise) ------------
    if (tid == 0) {
        const float e00 = ldsGram[0], e01 = ldsGram[1], e02 = ldsGram[2];
        const float e11 = ldsGram[3], e12 = ldsGram[4], e22 = ldsGram[5];
        const float E[3][3] = {{e00, e01, e02}, {e01, e11, e12}, {e02, e12, e22}};
        #pragma unroll
        for (int i = 0; i < 3; ++i) {
            const float rmax = fmaxf(E[i][0], fmaxf(E[i][1], E[i][2]));
            float en[3], emax = -__builtin_inff();
            #pragma unroll
            for (int j = 0; j < 3; ++j) { en[j] = rmax - E[i][j]; emax = fmaxf(emax, en[j]); }
            float p[3], s = 0.0f;
            #pragma unroll
            for (int j = 0; j < 3; ++j) { p[j] = __expf(en[j] - emax); s += p[j]; }
            const float inv = 1.0f / s;
            #pragma unroll
            for (int j = 0; j < 3; ++j) ldsAtt[i * 3 + j] = p[j] * inv;
        }
    }
    __syncthreads();

    // ---- Phase 3: out = gamma * (attn V) + x, streamed, NT stores --------
    const float g = gamma[0];
    const float a00 = ldsAtt[0], a01 = ldsAtt[1], a02 = ldsAtt[2];
    const float a10 = ldsAtt[3], a11 = ldsAtt[4], a12 = ldsAtt[5];
    const float a20 = ldsAtt[6], a21 = ldsAtt[7], a22 = ldsAtt[8];

    #pragma unroll
    for (int it = 0; it < 4; ++it) {
        const int n = (tid + it * 256) * 4;       // 4 columns per b128
        const v4f v0 = *(const v4f*)&ldsV[0 * HW + n];
        const v4f v1 = *(const v4f*)&ldsV[1 * HW + n];
        const v4f v2 = *(const v4f*)&ldsV[2 * HW + n];
        const v4f o0 = g * (a00 * v0 + a01 * v1 + a02 * v2) + v0;
        const v4f o1 = g * (a10 * v0 + a11 * v1 + a12 * v2) + v1;
        const v4f o2 = g * (a20 * v0 + a21 * v1 + a22 * v2) + v2;
        __builtin_nontemporal_store(o0, (v4f*)&out[base + 0 * HW + n]);
        __builtin_nontemporal_store(o1, (v4f*)&out[base + 1 * HW + n]);
        __builtin_nontemporal_store(o2, (v4f*)&out[base + 2 * HW + n]);
    }
}

extern "C" void kernel_launch(void* const* d_in, const int* in_sizes, int n_in,
                              void* d_out, int out_size, void* d_ws, size_t ws_size,
                              hipStream_t stream) {
    const float* x     = (const float*)d_in[0];
    const float* gamma = (const float*)d_in[1];
    float* out         = (float*)d_out;

    const int nbc = in_sizes[0] / TILE;          // B*C = 2048 (b,c) tiles
    cam_fused_kernel<<<nbc, 256, 0, stream>>>(x, gamma, out);
}